// Model_3925600108806
// MI455X (gfx1250) — compile-verified
//
#include <hip/hip_runtime.h>

typedef __attribute__((ext_vector_type(2))) float v2f;
typedef __attribute__((ext_vector_type(8))) float v8f;

#define B_ROWS 16384
#define DIM    256
#define K_IMG  2048
#define F_ATT  40

__device__ __forceinline__ v8f wmma_f32(v2f a, v2f b, v8f c) {
  // 8 args: (neg_a, A, neg_b, B, c_mod, C, reuse_a, reuse_b)
  return __builtin_amdgcn_wmma_f32_16x16x4_f32(
      false, a, false, b, (short)0, c, false, false);
}

// Async global -> LDS copy, 16 bytes per lane, tracked by ASYNCcnt.
__device__ __forceinline__ void async_copy_b128(unsigned lds_addr,
                                                const float* gptr) {
  asm volatile("global_load_async_to_lds_b128 %0, %1, off"
               :
               : "v"(lds_addr), "v"(gptr)
               : "memory");
}

template <int N>
__device__ __forceinline__ void wait_asynccnt() {
  asm volatile("s_wait_asynccnt %0" : : "n"(N) : "memory");
}

// ---------------------------------------------------------------------------
// K0: zero BN stat accumulators + correct counter
// ---------------------------------------------------------------------------
__global__ void zero_ws_kernel(float* sums, float* sumsq, int* count) {
  int i = threadIdx.x;
  if (i < 2 * DIM) { sums[i] = 0.0f; sumsq[i] = 0.0f; }
  if (i == 0) *count = 0;
}

// ---------------------------------------------------------------------------
// K1: emb_raw[16384 x 256] = img[16384 x 2048] @ W_img + b_img
// Block tile 128x64, K-chunk 32, 8 waves each computing 32x32 (2x2 WMMA
// tiles). Double-buffered LDS fed by GLOBAL_LOAD_ASYNC_TO_LDS_B128:
// chunk i+1 streams into buf^1 while WMMA consumes buf.
//   As: [buf][128 rows][36 floats]  (pad 36 -> 16B-aligned rows, bank-safe b64)
//   Bs: [buf][32 rows][72 floats]   (untransposed; pad 72 -> disjoint bank sets)
// ---------------------------------------------------------------------------
__global__ __launch_bounds__(256) void gemm1_async_kernel(
    const float* __restrict__ A, const float* __restrict__ W,
    const float* __restrict__ bias, float* __restrict__ C) {
  constexpr int ASZ = 128 * 36 * 4;  // bytes per A buffer
  constexpr int BSZ = 32 * 72 * 4;   // bytes per B buffer
  __shared__ __align__(16) float As[2][128 * 36];
  __shared__ __align__(16) float Bs[2][32 * 72];

  const int tid  = threadIdx.x;
  const int n0   = blockIdx.x * 64;
  const int m0   = blockIdx.y * 128;
  const int lane = tid & 31;
  const int w    = tid >> 5;
  const int wm   = (w & 3) * 32;
  const int wn   = (w >> 2) * 32;
  const int hl   = lane >> 4;
  const int l15  = lane & 15;

  // per-thread async-copy descriptors (element offsets)
  int aoff[4]; unsigned alds[4];
#pragma unroll
  for (int j = 0; j < 4; ++j) {
    int i = tid + 256 * j;
    int r = i >> 3, c4 = (i & 7) << 2;
    aoff[j] = r * K_IMG + c4;
    alds[j] = (unsigned)(r * 36 + c4);
  }
  int boff[2]; unsigned blds[2];
#pragma unroll
  for (int j = 0; j < 2; ++j) {
    int i = tid + 256 * j;
    int r = i >> 4, c4 = (i & 15) << 2;
    boff[j] = r * DIM + c4;
    blds[j] = (unsigned)(r * 72 + c4);
  }

  const unsigned asBase = (unsigned)(uintptr_t)&As[0][0];
  const unsigned bsBase = (unsigned)(uintptr_t)&Bs[0][0];

  auto issue = [&](int chunk, int buf) {
    const float* Ag = A + (long)m0 * K_IMG + chunk * 32;
    const float* Bg = W + (long)chunk * 32 * DIM + n0;
#pragma unroll
    for (int j = 0; j < 4; ++j)
      async_copy_b128(asBase + (unsigned)buf * ASZ + alds[j] * 4u, Ag + aoff[j]);
#pragma unroll
    for (int j = 0; j < 2; ++j)
      async_copy_b128(bsBase + (unsigned)buf * BSZ + blds[j] * 4u, Bg + boff[j]);
  };

  v8f acc[2][2];
#pragma unroll
  for (int i = 0; i < 2; ++i)
#pragma unroll
    for (int j = 0; j < 2; ++j)
#pragma unroll
      for (int q = 0; q < 8; ++q) acc[i][j][q] = 0.0f;

  constexpr int NIT = K_IMG / 32;
  issue(0, 0);  // prologue: chunk 0 -> buffer 0

  for (int it = 0; it < NIT; ++it) {
    const int buf = it & 1;
    if (it + 1 < NIT) {
      issue(it + 1, buf ^ 1);  // prefetch next chunk into other buffer
      wait_asynccnt<6>();      // retire current chunk's 6 copies (in-order)
    } else {
      wait_asynccnt<0>();
    }
    __syncthreads();  // current buffer fully populated by all waves

    const float* as = &As[buf][0];
    const float* bs = &Bs[buf][0];
#pragma unroll
    for (int kk = 0; kk < 8; ++kk) {
      const int kidx = kk * 4 + hl * 2;
      v2f a0 = *(const v2f*)(as + (wm + l15) * 36 + kidx);
      v2f a1 = *(const v2f*)(as + (wm + 16 + l15) * 36 + kidx);
      v2f b0, b1;
      b0.x = bs[kidx * 72 + wn + l15];
      b0.y = bs[(kidx + 1) * 72 + wn + l15];
      b1.x = bs[kidx * 72 + wn + 16 + l15];
      b1.y = bs[(kidx + 1) * 72 + wn + 16 + l15];
      acc[0][0] = wmma_f32(a0, b0, acc[0][0]);
      acc[0][1] = wmma_f32(a0, b1, acc[0][1]);
      acc[1][0] = wmma_f32(a1, b0, acc[1][0]);
      acc[1][1] = wmma_f32(a1, b1, acc[1][1]);
    }
    __syncthreads();  // everyone done reading buf before it is refilled
  }

  // epilogue: + bias, store
#pragma unroll
  for (int mt = 0; mt < 2; ++mt)
#pragma unroll
    for (int nt = 0; nt < 2; ++nt) {
      const int ncol  = n0 + wn + nt * 16 + l15;
      const float bv  = bias[ncol];
      const int mbase = m0 + wm + mt * 16 + hl * 8;
#pragma unroll
      for (int r = 0; r < 8; ++r)
        C[(long)(mbase + r) * DIM + ncol] = acc[mt][nt][r] + bv;
    }
}

// ---------------------------------------------------------------------------
// K4: e_raw = relu(bn(h)) @ W2 + b2   (K=256, BN affine+ReLU fused on A load)
// Same tiling as round 1 (VGPR-staged: the BN transform must touch VGPRs).
// gridDim.z = 2 selects the pos/neg branch.
// ---------------------------------------------------------------------------
__global__ __launch_bounds__(256) void gemm2_bn_kernel(
    const float* __restrict__ Aall, const float* __restrict__ W,
    const float* __restrict__ bias, float* __restrict__ Call,
    long strideA, long strideC, const float* __restrict__ scaleAll,
    const float* __restrict__ shiftAll) {
  __shared__ float As[128][36];
  __shared__ float Bs[64][36];  // Bs[n][k] (transposed W tile)

  const int z = blockIdx.z;
  const float* A = Aall + (long)z * strideA;
  float* C = Call + (long)z * strideC;
  const float* scale = scaleAll + z * DIM;
  const float* shift = shiftAll + z * DIM;

  const int tid  = threadIdx.x;
  const int n0   = blockIdx.x * 64;
  const int m0   = blockIdx.y * 128;
  const int lane = tid & 31;
  const int w    = tid >> 5;
  const int wm   = (w & 3) * 32;
  const int wn   = (w >> 2) * 32;
  const int hl   = lane >> 4;
  const int l15  = lane & 15;

  v8f acc[2][2];
#pragma unroll
  for (int i = 0; i < 2; ++i)
#pragma unroll
    for (int j = 0; j < 2; ++j)
#pragma unroll
      for (int q = 0; q < 8; ++q) acc[i][j][q] = 0.0f;

  for (int k0 = 0; k0 < DIM; k0 += 32) {
    __syncthreads();
#pragma unroll
    for (int j = 0; j < 4; ++j) {
      int i  = tid + 256 * j;
      int r  = i >> 3;
      int c4 = (i & 7) << 2;
      float4 v = *(const float4*)(A + (long)(m0 + r) * DIM + k0 + c4);
      v.x = fmaxf(fmaf(v.x, scale[k0 + c4 + 0], shift[k0 + c4 + 0]), 0.0f);
      v.y = fmaxf(fmaf(v.y, scale[k0 + c4 + 1], shift[k0 + c4 + 1]), 0.0f);
      v.z = fmaxf(fmaf(v.z, scale[k0 + c4 + 2], shift[k0 + c4 + 2]), 0.0f);
      v.w = fmaxf(fmaf(v.w, scale[k0 + c4 + 3], shift[k0 + c4 + 3]), 0.0f);
      *(float4*)&As[r][c4] = v;
    }
#pragma unroll
    for (int j = 0; j < 2; ++j) {
      int i  = tid + 256 * j;
      int r  = i >> 4;
      int c4 = (i & 15) << 2;
      float4 v = *(const float4*)(W + (long)(k0 + r) * DIM + n0 + c4);
      Bs[c4 + 0][r] = v.x;
      Bs[c4 + 1][r] = v.y;
      Bs[c4 + 2][r] = v.z;
      Bs[c4 + 3][r] = v.w;
    }
    __syncthreads();
#pragma unroll
    for (int kk = 0; kk < 8; ++kk) {
      const int kidx = kk * 4 + hl * 2;
      v2f a0 = *(const v2f*)&As[wm + l15][kidx];
      v2f a1 = *(const v2f*)&As[wm + 16 + l15][kidx];
      v2f b0 = *(const v2f*)&Bs[wn + l15][kidx];
      v2f b1 = *(const v2f*)&Bs[wn + 16 + l15][kidx];
      acc[0][0] = wmma_f32(a0, b0, acc[0][0]);
      acc[0][1] = wmma_f32(a0, b1, acc[0][1]);
      acc[1][0] = wmma_f32(a1, b0, acc[1][0]);
      acc[1][1] = wmma_f32(a1, b1, acc[1][1]);
    }
  }

#pragma unroll
  for (int mt = 0; mt < 2; ++mt)
#pragma unroll
    for (int nt = 0; nt < 2; ++nt) {
      const int ncol  = n0 + wn + nt * 16 + l15;
      const float bv  = bias[ncol];
      const int mbase = m0 + wm + mt * 16 + hl * 8;
#pragma unroll
      for (int r = 0; r < 8; ++r)
        C[(long)(mbase + r) * DIM + ncol] = acc[mt][nt][r] + bv;
    }
}

// ---------------------------------------------------------------------------
// K2: h = att @ W1 + b1 (K=40) for both branches; accumulate column stats.
// ---------------------------------------------------------------------------
__global__ __launch_bounds__(256) void att_fc1_kernel(
    const float* __restrict__ attP, const float* __restrict__ attN,
    const float* __restrict__ W1, const float* __restrict__ b1,
    float* __restrict__ hOut, float* __restrict__ sums,
    float* __restrict__ sumsq) {
  __shared__ float aL[64][F_ATT];
  const int z = blockIdx.y;
  const float* att = z ? attN : attP;
  float* h = hOut + (long)z * B_ROWS * DIM;
  const int r0  = blockIdx.x * 64;
  const int tid = threadIdx.x;

  for (int i = tid; i < 64 * F_ATT; i += 256) {
    int r = i / F_ATT, c = i % F_ATT;
    aL[r][c] = att[(long)(r0 + r) * F_ATT + c];
  }
  __syncthreads();

  float wr[F_ATT];
#pragma unroll
  for (int k = 0; k < F_ATT; ++k) wr[k] = W1[k * DIM + tid];
  const float bv = b1[tid];

  float s = 0.0f, s2 = 0.0f;
  for (int r = 0; r < 64; ++r) {
    float acc = bv;
#pragma unroll
    for (int k = 0; k < F_ATT; ++k) acc = fmaf(aL[r][k], wr[k], acc);
    h[(long)(r0 + r) * DIM + tid] = acc;
    s += acc;
    s2 += acc * acc;
  }
  atomicAdd(&sums[z * DIM + tid], s);
  atomicAdd(&sumsq[z * DIM + tid], s2);
}

// ---------------------------------------------------------------------------
// K3: fold BatchNorm (batch stats, biased var) into per-feature scale/shift
// ---------------------------------------------------------------------------
__global__ void bn_finalize_kernel(const float* __restrict__ sums,
                                   const float* __restrict__ sumsq,
                                   const float* __restrict__ gamma,
                                   const float* __restrict__ beta,
                                   float* __restrict__ scale,
                                   float* __restrict__ shift) {
  int i = blockIdx.x * blockDim.x + threadIdx.x;
  if (i < 2 * DIM) {
    int c = i & (DIM - 1);
    float mean = sums[i] * (1.0f / B_ROWS);
    float var  = sumsq[i] * (1.0f / B_ROWS) - mean * mean;
    float g    = gamma[c] * rsqrtf(var + 1e-3f);
    scale[i]   = g;
    shift[i]   = beta[c] - mean * g;
  }
}

// ---------------------------------------------------------------------------
// K5: per-row L2-normalize emb/ep/en (in place in d_out), pairwise distance
// (torch eps inside the abs), margin count. One wave per row.
// ---------------------------------------------------------------------------
__device__ __forceinline__ float wave_sum(float v) {
#pragma unroll
  for (int m = 16; m >= 1; m >>= 1) v += __shfl_xor(v, m, 32);
  return v;
}
__device__ __forceinline__ float dot4(float4 a) {
  return a.x * a.x + a.y * a.y + a.z * a.z + a.w * a.w;
}
__device__ __forceinline__ float pd4(float4 a, float4 b) {
  float dx = fabsf(a.x - b.x + 1e-6f);
  float dy = fabsf(a.y - b.y + 1e-6f);
  float dz = fabsf(a.z - b.z + 1e-6f);
  float dw = fabsf(a.w - b.w + 1e-6f);
  return dx * dx + dy * dy + dz * dz + dw * dw;
}
__device__ __forceinline__ void scale4(float4& a, float s) {
  a.x *= s; a.y *= s; a.z *= s; a.w *= s;
}

__global__ __launch_bounds__(256) void norm_dist_kernel(float* __restrict__ out,
                                                        int* __restrict__ count) {
  const int lane = threadIdx.x & 31;
  const int wv   = threadIdx.x >> 5;
  const long row = (long)blockIdx.x * 8 + wv;
  float* e = out + row * DIM;
  float* p = out + (long)B_ROWS * DIM + row * DIM;
  float* q = out + 2L * B_ROWS * DIM + row * DIM;

  float4 ev0 = *(float4*)(e + 4 * lane), ev1 = *(float4*)(e + 128 + 4 * lane);
  float4 pv0 = *(float4*)(p + 4 * lane), pv1 = *(float4*)(p + 128 + 4 * lane);
  float4 qv0 = *(float4*)(q + 4 * lane), qv1 = *(float4*)(q + 128 + 4 * lane);

  float ie = 1.0f / sqrtf(wave_sum(dot4(ev0) + dot4(ev1)));
  float ip = 1.0f / sqrtf(wave_sum(dot4(pv0) + dot4(pv1)));
  float iq = 1.0f / sqrtf(wave_sum(dot4(qv0) + dot4(qv1)));

  scale4(ev0, ie); scale4(ev1, ie);
  scale4(pv0, ip); scale4(pv1, ip);
  scale4(qv0, iq); scale4(qv1, iq);

  *(float4*)(e + 4 * lane) = ev0; *(float4*)(e + 128 + 4 * lane) = ev1;
  *(float4*)(p + 4 * lane) = pv0; *(float4*)(p + 128 + 4 * lane) = pv1;
  *(float4*)(q + 4 * lane) = qv0; *(float4*)(q + 128 + 4 * lane) = qv1;

  float dp = sqrtf(wave_sum(pd4(ev0, pv0) + pd4(ev1, pv1)));
  float dn = sqrtf(wave_sum(pd4(ev0, qv0) + pd4(ev1, qv1)));
  if (lane == 0 && (dn - dp) > 0.3f) atomicAdd(count, 1);
}

// ---------------------------------------------------------------------------
// Launcher
// ---------------------------------------------------------------------------
extern "C" void kernel_launch(void* const* d_in, const int* in_sizes, int n_in,
                              void* d_out, int out_size, void* d_ws, size_t ws_size,
                              hipStream_t stream) {
  const float* img   = (const float*)d_in[0];
  const float* att_p = (const float*)d_in[1];
  const float* att_n = (const float*)d_in[2];
  const float* W_img = (const float*)d_in[3];
  const float* b_img = (const float*)d_in[4];
  const float* W1    = (const float*)d_in[5];
  const float* b1    = (const float*)d_in[6];
  const float* gamma = (const float*)d_in[7];
  const float* beta  = (const float*)d_in[8];
  const float* W2    = (const float*)d_in[9];
  const float* b2    = (const float*)d_in[10];

  float* out = (float*)d_out;
  const long BD = (long)B_ROWS * DIM;

  // workspace layout (floats)
  float* ws    = (float*)d_ws;
  float* h     = ws;                  // 2 * B * D
  float* sums  = h + 2 * BD;          // 2 * 256
  float* sumsq = sums + 2 * DIM;      // 2 * 256
  float* scale = sumsq + 2 * DIM;     // 2 * 256
  float* shift = scale + 2 * DIM;     // 2 * 256
  int* count   = (int*)(out + 3 * BD);

  // K0: zero accumulators + counter
  zero_ws_kernel<<<1, 512, 0, stream>>>(sums, sumsq, count);

  // K1: emb_raw = img @ W_img + b_img  -> d_out[0 : B*D]   (async double-buffer)
  gemm1_async_kernel<<<dim3(DIM / 64, B_ROWS / 128, 1), 256, 0, stream>>>(
      img, W_img, b_img, out);

  // K2: h = att @ W1 + b1 (both branches) + column stats
  att_fc1_kernel<<<dim3(B_ROWS / 64, 2), 256, 0, stream>>>(att_p, att_n, W1, b1,
                                                           h, sums, sumsq);

  // K3: BN fold
  bn_finalize_kernel<<<2, 256, 0, stream>>>(sums, sumsq, gamma, beta, scale, shift);

  // K4: e_raw = relu(bn(h)) @ W2 + b2  -> d_out[B*D : 3*B*D]
  gemm2_bn_kernel<<<dim3(DIM / 64, B_ROWS / 128, 2), 256, 0, stream>>>(
      h, W2, b2, out + BD, BD, BD, scale, shift);

  // K5: normalize + distances + margin count
  norm_dist_kernel<<<B_ROWS / 8, 256, 0, stream>>>(out, count);
}